// MessageLayer_52407190946315
// MI455X (gfx1250) — compile-verified
//
#include <hip/hip_runtime.h>

typedef float v2f __attribute__((ext_vector_type(2)));
typedef float v8f __attribute__((ext_vector_type(8)));

#define WAVES 8
#define NPASS 4

static __device__ __forceinline__ v2f mkv2(float x, float y) {
  v2f r; r.x = x; r.y = y; return r;
}

static __device__ __forceinline__ v8f wmma4(v2f a, v2f b, v8f c) {
  // D = A(16x4 f32) * B(4x16 f32) + C(16x16 f32)
  return __builtin_amdgcn_wmma_f32_16x16x4_f32(false, a, false, b, (short)0, c,
                                               false, false);
}

static __device__ __forceinline__ float comp3(float3 a, int k) {
  return k == 0 ? a.x : (k == 1 ? a.y : a.z);
}

static __device__ __forceinline__ float ldnt(const float* p) {
  return __builtin_nontemporal_load(p);
}
static __device__ __forceinline__ int ldnti(const int* p) {
  return __builtin_nontemporal_load(p);
}

__global__ void __launch_bounds__(256)
msg_zero(float* __restrict__ out, int n) {
  int i = blockIdx.x * blockDim.x + threadIdx.x;
  if (i < n) out[i] = 0.f;
}

__global__ void __launch_bounds__(WAVES * 32)
msg_edge(const float* __restrict__ node, const int* __restrict__ ei,
         const float* __restrict__ sh0, const float* __restrict__ sh1,
         const float* __restrict__ sh2, const float* __restrict__ W,
         float* __restrict__ out, int E) {
  const int lane = threadIdx.x & 31;
  const int wave = threadIdx.x >> 5;
  const int h = lane >> 4;      // lane half: supplies K rows {4c+2h, 4c+2h+1}
  const int o = lane & 15;      // B/D column
  const bool ocol = (o < 8);

  // ---- constant weight operands (B matrices), built once per wave ----
  // out[o] += W[idx][i][o] * base[i];  W flat: idx*64 + i*8 + o
  // b_v : vec-path rows (K blocks 011,101,111,211), outputs in cols 0-7
  // b_sc: scalar-path rows (K blocks 000,110),      outputs in cols 8-15
  // The fused k=0 matmul uses B = [b_v(rows 0-31); b_sc(rows 32-47)] so that
  // all 16 D columns carry real outputs.
  v2f b_v[8], b_sc[4];
  {
    const int r2 = 2 * h;
    const int cmap[4] = {1, 2, 4, 5};
#pragma unroll
    for (int c = 0; c < 8; ++c) {
      int r0 = 4 * c + r2, r1 = r0 + 1;
      b_v[c].x = ocol ? W[cmap[r0 >> 3] * 64 + (r0 & 7) * 8 + o] : 0.f;
      b_v[c].y = ocol ? W[cmap[r1 >> 3] * 64 + (r1 & 7) * 8 + o] : 0.f;
    }
#pragma unroll
    for (int c = 0; c < 4; ++c) {
      int r0 = 4 * c + r2, r1 = r0 + 1;
      b_sc[c].x = !ocol ? W[((r0 < 8) ? 0 : 3) * 64 + (r0 & 7) * 8 + (o - 8)] : 0.f;
      b_sc[c].y = !ocol ? W[((r1 < 8) ? 0 : 3) * 64 + (r1 & 7) * 8 + (o - 8)] : 0.f;
    }
  }

  const float is3 = 0.57735026918962576f;    // 1/sqrt(3)
  const float is6 = 0.40824829046386302f;    // 1/sqrt(6)
  const float k1  = 0.31622776601683794f;    // 1/sqrt(10)
  const float k3  = 0.18257418583505536f;    // 1/sqrt(30)

  const int base0 = (blockIdx.x * WAVES + wave) * (NPASS * 16);

  for (int p = 0; p < NPASS; ++p) {
    const int e = base0 + p * 16 + o;        // lanes l and l+16 share edge (l&15)
    const bool valid = (e < E);

    int tgt = 0;                              // invalid edges: harmless 0-adds to node 0
    float s0 = 0, s1 = 0, s2 = 0, s3 = 0;
    float3 v[4] = {};
    float g0 = 0, g1x = 0, g1y = 0, g1z = 0;
    float g2a = 0, g2b = 0, g2c = 0, g2d = 0, g2e = 0;

    if (valid) {
      const int src = ldnti(ei + e);
      tgt = ldnti(ei + E + e);
      const float* nf = node + (size_t)src * 32;
      // this lane's 4 channels: {2h, 2h+1, 2h+4, 2h+5}
      const int cA = 2 * h;
      s0 = nf[cA]; s1 = nf[cA + 1]; s2 = nf[cA + 4]; s3 = nf[cA + 5];
      const int vb = 8 + 6 * h;              // v[c][k] = nf[8 + 3c + k]
      v[0] = make_float3(nf[vb + 0],  nf[vb + 1],  nf[vb + 2]);
      v[1] = make_float3(nf[vb + 3],  nf[vb + 4],  nf[vb + 5]);
      v[2] = make_float3(nf[vb + 12], nf[vb + 13], nf[vb + 14]);
      v[3] = make_float3(nf[vb + 15], nf[vb + 16], nf[vb + 17]);
      g0  = ldnt(sh0 + e);
      g1x = ldnt(sh1 + 3 * e);     g1y = ldnt(sh1 + 3 * e + 1);
      g1z = ldnt(sh1 + 3 * e + 2);
      g2a = ldnt(sh2 + 5 * e);     g2b = ldnt(sh2 + 5 * e + 1);
      g2c = ldnt(sh2 + 5 * e + 2); g2d = ldnt(sh2 + 5 * e + 3);
      g2e = ldnt(sh2 + 5 * e + 4);
    }

    // ---- CG-base features for this lane's 4 channels ----
    float sc[4] = {s0, s1, s2, s3};
    float3 cr[4], t211[4], t011[4];
    float t110[4];
#pragma unroll
    for (int c = 0; c < 4; ++c) {
      float3 vv = v[c];
      // (1,1,0): dot / sqrt3
      t110[c] = is3 * (g1x * vv.x + g1y * vv.y + g1z * vv.z);
      // (1,1,1): cross(sh1, v) / sqrt6
      cr[c].x = is6 * (g1y * vv.z - g1z * vv.y);
      cr[c].y = is6 * (g1z * vv.x - g1x * vv.z);
      cr[c].z = is6 * (g1x * vv.y - g1y * vv.x);
      // (0,1,1): sh0 * v / sqrt3
      float gg = is3 * g0;
      t011[c].x = gg * vv.x; t011[c].y = gg * vv.y; t011[c].z = gg * vv.z;
      // (2,1,1): real CG, rows (xy,yz,z2,zx,x2-y2), basis order (y,z,x)
      t211[c].x = k1 * (g2a * vv.z + g2b * vv.y - g2e * vv.x) - k3 * g2c * vv.x;
      t211[c].y = k1 * (g2b * vv.x + g2d * vv.z) + 2.f * k3 * g2c * vv.y;
      t211[c].z = k1 * (g2a * vv.x + g2d * vv.y + g2e * vv.z) - k3 * g2c * vv.z;
    }

    // D row j (this lane) is edge 8h+j: fetch its scaled target offset
    const int toff = tgt * 32;
    int tj[8];
#pragma unroll
    for (int j = 0; j < 8; ++j) tj[j] = __shfl(toff, 8 * h + j, 32);

    // ---- fused matmul: vec k=0 (cols 0-7) + scalar path (cols 8-15) ----
    {
      v2f a;
      v8f acc = {0.f, 0.f, 0.f, 0.f, 0.f, 0.f, 0.f, 0.f};
      a = mkv2(t011[0].x, t011[1].x);           acc = wmma4(a, b_v[0], acc);
      a = mkv2(t011[2].x, t011[3].x);           acc = wmma4(a, b_v[1], acc);
      a = mkv2(is3*g1x*sc[0], is3*g1x*sc[1]);   acc = wmma4(a, b_v[2], acc);
      a = mkv2(is3*g1x*sc[2], is3*g1x*sc[3]);   acc = wmma4(a, b_v[3], acc);
      a = mkv2(cr[0].x, cr[1].x);               acc = wmma4(a, b_v[4], acc);
      a = mkv2(cr[2].x, cr[3].x);               acc = wmma4(a, b_v[5], acc);
      a = mkv2(t211[0].x, t211[1].x);           acc = wmma4(a, b_v[6], acc);
      a = mkv2(t211[2].x, t211[3].x);           acc = wmma4(a, b_v[7], acc);
      a = mkv2(g0 * sc[0], g0 * sc[1]);         acc = wmma4(a, b_sc[0], acc);
      a = mkv2(g0 * sc[2], g0 * sc[3]);         acc = wmma4(a, b_sc[1], acc);
      a = mkv2(t110[0], t110[1]);               acc = wmma4(a, b_sc[2], acc);
      a = mkv2(t110[2], t110[3]);               acc = wmma4(a, b_sc[3], acc);
      // col o<8 -> vec out channel 8+3o (k=0); col o>=8 -> scalar channel o-8
      const int ch = ocol ? (8 + 3 * o) : (o - 8);
#pragma unroll
      for (int j = 0; j < 8; ++j)
        atomicAdd(out + (unsigned)(tj[j] + ch), acc[j]);   // all 32 lanes live
    }

    // ---- vec components k=1,2: cols 0-7 only ----
#pragma unroll
    for (int k = 1; k < 3; ++k) {
      const float shk = (k == 1) ? g1y : g1z;
      v2f a;
      v8f acc = {0.f, 0.f, 0.f, 0.f, 0.f, 0.f, 0.f, 0.f};
      a = mkv2(comp3(t011[0], k), comp3(t011[1], k));   acc = wmma4(a, b_v[0], acc);
      a = mkv2(comp3(t011[2], k), comp3(t011[3], k));   acc = wmma4(a, b_v[1], acc);
      a = mkv2(is3 * shk * sc[0], is3 * shk * sc[1]);   acc = wmma4(a, b_v[2], acc);
      a = mkv2(is3 * shk * sc[2], is3 * shk * sc[3]);   acc = wmma4(a, b_v[3], acc);
      a = mkv2(comp3(cr[0], k), comp3(cr[1], k));       acc = wmma4(a, b_v[4], acc);
      a = mkv2(comp3(cr[2], k), comp3(cr[3], k));       acc = wmma4(a, b_v[5], acc);
      a = mkv2(comp3(t211[0], k), comp3(t211[1], k));   acc = wmma4(a, b_v[6], acc);
      a = mkv2(comp3(t211[2], k), comp3(t211[3], k));   acc = wmma4(a, b_v[7], acc);
      const int ch = 8 + 3 * o + k;
      if (ocol) {                                // one branch per matmul
#pragma unroll
        for (int j = 0; j < 8; ++j)
          atomicAdd(out + (unsigned)(tj[j] + ch), acc[j]);
      }
    }
  }
}

extern "C" void kernel_launch(void* const* d_in, const int* in_sizes, int n_in,
                              void* d_out, int out_size, void* d_ws,
                              size_t ws_size, hipStream_t stream) {
  (void)n_in; (void)d_ws; (void)ws_size;
  const float* node = (const float*)d_in[0];
  const int*   ei   = (const int*)d_in[1];
  const float* sh0  = (const float*)d_in[2];
  const float* sh1  = (const float*)d_in[3];
  const float* sh2  = (const float*)d_in[4];
  const float* W    = (const float*)d_in[5];
  float* out = (float*)d_out;

  const int E = in_sizes[1] / 2;

  // zero the accumulation buffer every call (deterministic; graph-safe)
  msg_zero<<<(out_size + 255) / 256, 256, 0, stream>>>(out, out_size);

  const int edgesPerBlock = WAVES * NPASS * 16;  // 512
  const int blocks = (E + edgesPerBlock - 1) / edgesPerBlock;
  msg_edge<<<blocks, WAVES * 32, 0, stream>>>(node, ei, sh0, sh1, sh2, W, out, E);
}